// LIF_15504831939372
// MI455X (gfx1250) — compile-verified
//
#include <hip/hip_runtime.h>
#include <cstdint>
#include <cstddef>

// Problem constants (match reference)
#define Bc 8
#define Sc 64
#define Hc 256
#define Nc 256
#define Tc (Sc * Hc)   // 16384 scan steps

typedef float v2f __attribute__((ext_vector_type(2)));
typedef float v8f __attribute__((ext_vector_type(8)));

// ---------------------------------------------------------------------------
// CDNA5 async-tensor path: GLOBAL_LOAD_ASYNC_TO_LDS_B128 + S_WAIT_ASYNCCNT.
// ---------------------------------------------------------------------------
__device__ __forceinline__ void async_copy_b128(uint32_t lds_byte_addr, const void* gptr) {
    asm volatile("global_load_async_to_lds_b128 %0, %1, off"
                 :: "v"(lds_byte_addr), "v"(gptr) : "memory");
}
__device__ __forceinline__ void wait_async_0()  { asm volatile("s_wait_asynccnt 0"  ::: "memory"); }
__device__ __forceinline__ void wait_async_32() { asm volatile("s_wait_asynccnt 32" ::: "memory"); }

// Cheap sigmoid: v_mul + v_exp_f32 + v_add + v_rcp_f32 (no IEEE div fixup chain)
__device__ __forceinline__ float sigmoid_fast(float x) {
    return __builtin_amdgcn_rcpf(1.0f + __expf(-x));
}

// ---------------------------------------------------------------------------
// Phase 1a: sx[t*8 + b] = sigmoid(inputs[b*T + t]).  (512 KB, negligible)
// ---------------------------------------------------------------------------
__global__ void __launch_bounds__(256)
lif_sigmoid_transpose(const float* __restrict__ inp,
                      float* __restrict__ sx) {
    int idx = blockIdx.x * 256 + threadIdx.x;     // over B*T = 131072 exactly
    int b = idx / Tc;
    int t = idx - b * Tc;
    sx[t * Bc + b] = sigmoid_fast(inp[idx]);
}

// ---------------------------------------------------------------------------
// Phase 1b: the serial scan. 8 blocks x 1 wave32; lane <-> neuron.
// Async double-buffered LDS staging of sx; only the 16 MB state trajectory is
// written (stateT[n*T + t] = incoming carry at step t), batched as b128.
//   m = thresh - state;  y = max_b( x_b <= m ? x_b : -1 )
//   state' = (y < 0) ? 0 : state + y      // all-spike -> reset, matches ref
// ---------------------------------------------------------------------------
#define CHUNK 512                       // 512 steps * 8 * 4B = 16 KB per chunk
#define NCHUNK (Tc / CHUNK)             // 32
#define COPIES_PER_CHUNK ((CHUNK * Bc * 4) / (32 * 16))   // 32 b128 issues/wave

__device__ __forceinline__ void issue_chunk(const float* __restrict__ sx,
                                            int c, uint32_t lds_base, int lane) {
    const char* g = (const char*)(sx + (size_t)c * CHUNK * Bc);
#pragma unroll
    for (int j = 0; j < COPIES_PER_CHUNK; ++j) {
        uint32_t off = (uint32_t)((j * 32 + lane) * 16);
        async_copy_b128(lds_base + off, g + off);
    }
}

__global__ void __launch_bounds__(32)
lif_scan(const float* __restrict__ sx,
         const float* __restrict__ thresh,
         const float* __restrict__ acc0,
         float* __restrict__ stateT) {
    __shared__ __align__(16) float sbuf[2][CHUNK * Bc];   // 2 x 16 KB

    const int lane = threadIdx.x;
    const int n    = blockIdx.x * 32 + lane;
    float state    = acc0[n];
    const float th = thresh[n];
    float* __restrict__ outp = stateT + (size_t)n * Tc;   // 64KB-aligned per lane

    const uint32_t lds0 = (uint32_t)(uintptr_t)&sbuf[0][0];
    const uint32_t lds1 = (uint32_t)(uintptr_t)&sbuf[1][0];

    issue_chunk(sx, 0, lds0, lane);

    for (int c = 0; c < NCHUNK; ++c) {
        const int buf = c & 1;
        if (c + 1 < NCHUNK) {
            issue_chunk(sx, c + 1, ((c + 1) & 1) ? lds1 : lds0, lane);
            wait_async_32();          // newest 32 outstanding => chunk c landed
        } else {
            wait_async_0();
        }
        const float4* __restrict__ p4 = (const float4*)&sbuf[buf][0];
        const int tbase = c * CHUNK;

        auto step = [&](int i) {
            float4 a = p4[i * 2];
            float4 b = p4[i * 2 + 1];
            const float m = th - state;
            float c0 = (a.x <= m) ? a.x : -1.0f;
            float c1 = (a.y <= m) ? a.y : -1.0f;
            float c2 = (a.z <= m) ? a.z : -1.0f;
            float c3 = (a.w <= m) ? a.w : -1.0f;
            float c4 = (b.x <= m) ? b.x : -1.0f;
            float c5 = (b.y <= m) ? b.y : -1.0f;
            float c6 = (b.z <= m) ? b.z : -1.0f;
            float c7 = (b.w <= m) ? b.w : -1.0f;
            float y = fmaxf(fmaxf(fmaxf(c0, c1), fmaxf(c2, c3)),
                            fmaxf(fmaxf(c4, c5), fmaxf(c6, c7)));
            state = (y < 0.0f) ? 0.0f : (state + y);
        };

#pragma unroll 2
        for (int i = 0; i < CHUNK; i += 4) {     // 4 steps -> one b128 store
            float4 sv;
            sv.x = state; step(i + 0);
            sv.y = state; step(i + 1);
            sv.z = state; step(i + 2);
            sv.w = state; step(i + 3);
            *(float4*)(outp + tbase + i) = sv;   // carries for t..t+3
        }
    }
}

// ---------------------------------------------------------------------------
// Phase 2: bandwidth phase. One block per (b,s); lane = h; loop over n.
// Fully coalesced reads/stores; per-lane register accumulator for sum over n.
// Writes norm[b,s,h] to workspace (no atomics -> deterministic).
// ---------------------------------------------------------------------------
__global__ void __launch_bounds__(256)
lif_expand(const float* __restrict__ sx,
           const float* __restrict__ thresh,
           const float* __restrict__ stateT,
           float* __restrict__ outputs,
           float* __restrict__ spikes,
           float* __restrict__ norm) {
    const int bx = blockIdx.x;            // 0 .. B*S-1
    const int b  = bx / Sc;
    const int s  = bx - b * Sc;
    const int h  = threadIdx.x;
    const int t  = s * Hc + h;

    const float xb = sx[t * Bc + b];      // one gather per lane, hoisted
    const float* __restrict__ stp = stateT + t;
    const size_t obase = (size_t)bx * Nc * Hc + h;

    float sumsq = 0.0f;
#pragma unroll 4
    for (int n = 0; n < Nc; ++n) {
        if (n + 8 < Nc)
            __builtin_prefetch(stp + (size_t)(n + 8) * Tc, 0, 1);  // global_prefetch_b8
        const float st  = stp[(size_t)n * Tc];
        const float thn = thresh[n];                 // uniform -> s_load via K$
        const float acc = st + xb;
        const bool  spk = acc > thn;
        const float outv = spk ? sigmoid_fast(acc) : 0.5f;  // sigmoid(0)=0.5
        outputs[obase + (size_t)n * Hc] = outv;
        spikes [obase + (size_t)n * Hc] = spk ? 1.0f : 0.0f;
        sumsq = __fmaf_rn(outv, outv, sumsq);
    }
    norm[bx * Hc + h] = __fsqrt_rn(sumsq) * (1.0f / Nc);
}

// ---------------------------------------------------------------------------
// Phase 3: deterministic loss reduction on the MATRIX pipe.
// Single block, 8 waves. A = ones(16x4) makes V_WMMA_F32_16X16X4_F32 compute
// D[m][j] = sum_k B[k][j] + C[m][j]: a 64-way add per instruction whose total
// is independent of the B operand layout (pure sum). Each wave folds 16384
// floats into one accumulator tile, then acc[0] is lane-reduced (rows are
// identical copies -> halve). Fixed association order => bit-deterministic.
// ---------------------------------------------------------------------------
__global__ void __launch_bounds__(256)
lif_loss_reduce(const float* __restrict__ norm,
                const int* __restrict__ dtype,
                float* __restrict__ loss) {
    const int wave = threadIdx.x >> 5;    // 0..7
    const int lane = threadIdx.x & 31;

#if __has_builtin(__builtin_amdgcn_wmma_f32_16x16x4_f32)
    v8f acc = {};
    v2f ones; ones.x = 1.0f; ones.y = 1.0f;
    const float* base = norm + wave * ((Bc * Tc) / 8);      // 16384 floats/wave
#pragma unroll 4
    for (int i = 0; i < (Bc * Tc) / 8 / 64; ++i) {          // 256 wmma per wave
        float2 d = ((const float2*)(base + i * 64))[lane];  // 64 values/wmma
        v2f bv; bv.x = d.x; bv.y = d.y;
        acc = __builtin_amdgcn_wmma_f32_16x16x4_f32(
            /*neg_a=*/false, ones, /*neg_b=*/false, bv,
            /*c_mod=*/(short)0, acc, /*reuse_a=*/false, /*reuse_b=*/false);
    }
    float v = acc[0];                       // lanes 0-15: colsums; 16-31: copy
    for (int off = 16; off > 0; off >>= 1) v += __shfl_xor(v, off, 32);
    v *= 0.5f;                              // rows M=0 and M=8 both counted
#else
    float v = 0.0f;
    const float* base = norm + wave * ((Bc * Tc) / 8);
    for (int i = lane; i < (Bc * Tc) / 8; i += 32) v += base[i];
    for (int off = 16; off > 0; off >>= 1) v += __shfl_xor(v, off, 32);
#endif

    __shared__ float wsum[8];
    if (lane == 0) wsum[wave] = v;
    __syncthreads();
    if (threadIdx.x == 0) {
        float s = 0.0f;
#pragma unroll
        for (int w = 0; w < 8; ++w) s += wsum[w];
        const float sign = (dtype[0] == 1) ? -1.0f : 1.0f;
        *loss = sign * s * (1.0f / ((float)Bc * Sc * Hc));
    }
}

// ---------------------------------------------------------------------------
// Launch. d_out = outputs(33.5M) ++ spikes(33.5M) ++ loss(1), all f32.
// Workspace: sx (T*8) | stateT (N*T) | norm (B*T)  ~= 17.8 MB.
// ---------------------------------------------------------------------------
extern "C" void kernel_launch(void* const* d_in, const int* in_sizes, int n_in,
                              void* d_out, int out_size, void* d_ws, size_t ws_size,
                              hipStream_t stream) {
    (void)in_sizes; (void)n_in; (void)out_size; (void)ws_size;
    const float* inp  = (const float*)d_in[0];
    const float* thr  = (const float*)d_in[1];
    const float* ac0  = (const float*)d_in[2];
    const int*   dt   = (const int*)d_in[3];

    float* out      = (float*)d_out;
    float* outputs  = out;
    float* spikes   = out + (size_t)Bc * Sc * Nc * Hc;
    float* loss     = out + 2 * (size_t)Bc * Sc * Nc * Hc;

    float* sx     = (float*)d_ws;                  // T*8 floats
    float* stateT = sx + (size_t)Tc * Bc;          // N*T floats
    float* norm   = stateT + (size_t)Nc * Tc;      // B*T floats (8B-aligned)

    hipLaunchKernelGGL(lif_sigmoid_transpose, dim3((Bc * Tc) / 256), dim3(256), 0, stream,
                       inp, sx);
    hipLaunchKernelGGL(lif_scan, dim3(Nc / 32), dim3(32), 0, stream,
                       sx, thr, ac0, stateT);
    hipLaunchKernelGGL(lif_expand, dim3(Bc * Sc), dim3(256), 0, stream,
                       sx, thr, stateT, outputs, spikes, norm);
    hipLaunchKernelGGL(lif_loss_reduce, dim3(1), dim3(256), 0, stream,
                       norm, dt, loss);
}